// GDSSMlayer_33097017983473
// MI455X (gfx1250) — compile-verified
//
#include <hip/hip_runtime.h>

#define LSEQ 4096
#define HDIM 128
#define DDIM 128

typedef __attribute__((ext_vector_type(2))) float v2f;
typedef __attribute__((ext_vector_type(8))) float v8f;

__device__ __forceinline__ float gelu_tanh(float x) {
    const float c = 0.7978845608028654f;  // sqrt(2/pi)
    float u = c * (x + 0.044715f * x * x * x);
    return 0.5f * x * (1.0f + tanhf(u));
}

__device__ __forceinline__ float wave_sum(float p) {
#pragma unroll
    for (int off = 16; off >= 1; off >>= 1)
        p += __shfl_xor(p, off, 32);
    return p;
}

// One block = one 16-step tile of the sequence. 8 waves:
//  phase 1: each wave handles 2 timesteps -> three fused matvecs + A@t2 + gelu -> G tile in LDS
//  phase 2: each wave computes one 16-col slab of G @ W2 with v_wmma_f32_16x16x4_f32,
//           then the sigmoid gate and the y write.
__global__ __launch_bounds__(256)
void gdssm_fused(const float* __restrict__ zs,
                 const float* __restrict__ os_,
                 const float* __restrict__ skip_zs1,
                 const float* __restrict__ skip_zs2,
                 const float* __restrict__ W2,
                 const float* __restrict__ b2,
                 float* __restrict__ out_zs,
                 float* __restrict__ out_os,
                 float* __restrict__ out_y)
{
    extern __shared__ float smem[];
    float* Alds = smem;                    // 128*128 = 16384 f32 (A = skip_zs1[L-1])
    float* Glds = smem + 16384;            // 16*128  =  2048 f32 (gelu outputs)
    float* sws  = smem + 16384 + 2048;     // 8*256   =  2048 f32 (per-wave scratch)

    const int tid  = threadIdx.x;
    const int lane = tid & 31;
    const int wid  = tid >> 5;
    const int tile = blockIdx.x;

    // Stage A into LDS once per block (hot in L2 across all 256 blocks).
    {
        const float4* A4  = (const float4*)(skip_zs1 + (size_t)(LSEQ - 1) * HDIM * HDIM);
        float4*       Al4 = (float4*)Alds;
#pragma unroll
        for (int j = 0; j < 16; ++j)
            Al4[tid + 256 * j] = A4[tid + 256 * j];
    }
    __syncthreads();

    float* yws = sws + wid * 256;          // [0:128) y accum, [128:256) t2

    for (int r = 0; r < 2; ++r) {
        const int lrow = wid * 2 + r;
        const int l    = tile * 16 + lrow;

        const float4* zr  = (const float4*)(zs       + (size_t)l * HDIM * HDIM);
        const float4* s1r = (const float4*)(skip_zs1 + (size_t)l * HDIM * HDIM);
        const float4* s2r = (const float4*)(skip_zs2 + (size_t)l * HDIM * HDIM);
        float4*       ozr = (float4*)(out_zs + (size_t)l * HDIM * HDIM);

        // v chunk: lane owns os_[l][4*lane .. 4*lane+3]; also pass os_ through to d_out.
        float4 v = ((const float4*)(os_ + (size_t)l * HDIM))[lane];
        ((float4*)(out_os + (size_t)l * HDIM))[lane] = v;

        // t13[i] = ((zs+skip_zs1)[l] @ os_[l])[i] ; t2[i] = (skip_zs2[l] @ os_[l])[i]
        for (int i = 0; i < HDIM; ++i) {
            float4 az = zr[i * 32 + lane];
            ozr[i * 32 + lane] = az;                       // fused zs pass-through copy
            float4 a1 = s1r[i * 32 + lane];
            float4 a2 = s2r[i * 32 + lane];
            float p = (az.x + a1.x) * v.x + (az.y + a1.y) * v.y
                    + (az.z + a1.z) * v.z + (az.w + a1.w) * v.w;
            float q = a2.x * v.x + a2.y * v.y + a2.z * v.z + a2.w * v.w;
            p = wave_sum(p);
            q = wave_sum(q);
            if (lane == 0) { yws[i] = p; yws[128 + i] = q; }
        }

        // y += A @ t2   (A from LDS)
        float4 t2c = *(const float4*)&yws[128 + lane * 4];
        for (int i = 0; i < HDIM; ++i) {
            float4 a4 = *(const float4*)&Alds[i * 128 + lane * 4];
            float pa = a4.x * t2c.x + a4.y * t2c.y + a4.z * t2c.z + a4.w * t2c.w;
            pa = wave_sum(pa);
            if (lane == 0) yws[i] += pa;
        }

        // g = gelu(y) -> G tile row
#pragma unroll
        for (int j = 0; j < 4; ++j) {
            int i = lane + 32 * j;
            Glds[lrow * 128 + i] = gelu_tanh(yws[i]);
        }
    }
    __syncthreads();

    // ---- WMMA phase: (16 x 128) G times (128 x 128) W2; wave wid owns cols [16*wid, 16*wid+16) ----
    const int nb = wid * 16;
    const int lm = lane & 15;      // M index for A-frag, N index for B-frag / C-frag
    const int hi = lane >> 4;      // lane half selects k pair {4s,4s+1} vs {4s+2,4s+3}
    v8f acc = {};
    for (int s = 0; s < 32; ++s) {
        int kA = s * 4 + 2 * hi;
        v2f a, b;
        a.x = Glds[lm * 128 + kA];                  // A[m][kA]   (VGPR0: K=0 / K=2)
        a.y = Glds[lm * 128 + kA + 1];              // A[m][kA+1] (VGPR1: K=1 / K=3)
        b.x = W2[(size_t)kA * DDIM + nb + lm];      // B[kA][n]
        b.y = W2[(size_t)(kA + 1) * DDIM + nb + lm];
        acc = __builtin_amdgcn_wmma_f32_16x16x4_f32(
            /*neg_a=*/false, a, /*neg_b=*/false, b,
            /*c_mod=*/(short)0, acc, /*reuse_a=*/false, /*reuse_b=*/false);
    }

    // C/D layout: VGPR j -> row (j + 8*hi), col lane%16. Gate + write y.
#pragma unroll
    for (int j = 0; j < 8; ++j) {
        int m = j + 8 * hi;
        int n = nb + lm;
        float pre = acc[j] + b2[n];
        float sig = 1.0f / (1.0f + __expf(-pre));
        float g   = Glds[m * 128 + n];
        out_y[(size_t)(tile * 16 + m) * DDIM + n] = g * sig;
    }
}

extern "C" void kernel_launch(void* const* d_in, const int* in_sizes, int n_in,
                              void* d_out, int out_size, void* d_ws, size_t ws_size,
                              hipStream_t stream) {
    // setup_inputs order: x(unused), zs, os_, skip_zs1, skip_zs2, W2, b2
    const float* zs  = (const float*)d_in[1];
    const float* os_ = (const float*)d_in[2];
    const float* s1  = (const float*)d_in[3];
    const float* s2  = (const float*)d_in[4];
    const float* W2  = (const float*)d_in[5];
    const float* b2  = (const float*)d_in[6];

    float* out_zs = (float*)d_out;                                // L*H*H
    float* out_os = out_zs + (size_t)LSEQ * HDIM * HDIM;          // L*H
    float* out_y  = out_os + (size_t)LSEQ * HDIM;                 // L*D

    const size_t shmem = (size_t)(16384 + 2048 + 2048) * sizeof(float);  // 80 KB
    hipLaunchKernelGGL(gdssm_fused, dim3(LSEQ / 16), dim3(256), shmem, stream,
                       zs, os_, s1, s2, W2, b2, out_zs, out_os, out_y);
}